// Fusion_Block_936302870541
// MI455X (gfx1250) — compile-verified
//
#include <hip/hip_runtime.h>

typedef __attribute__((ext_vector_type(2))) float v2f;
typedef __attribute__((ext_vector_type(4))) float v4f;
typedef __attribute__((ext_vector_type(8))) float v8f;

#define L_LEN 8192
#define BLOCK 256
#define PER_THREAD_VEC 8   // 8 x v4f = 32 floats per thread; 256*32 = 8192 = one row

#if __has_builtin(__builtin_amdgcn_wmma_f32_16x16x4_f32)
#define USE_WMMA 1
#else
#define USE_WMMA 0
#endif

// One workgroup per (b,c) row. out[b,c,l] = y[b,c,l] * (mean_x_row + mean_y_row).
// Streams x and y exactly once (non-temporal), keeps y in VGPRs, streams out.
__global__ __launch_bounds__(BLOCK) void rowscale_kernel(
    const float* __restrict__ x, const float* __restrict__ y,
    float* __restrict__ out)
{
    const int tid = threadIdx.x;
    const long long rowbase = (long long)blockIdx.x * L_LEN;
    const v4f* xv4 = reinterpret_cast<const v4f*>(x + rowbase);
    const v4f* yv4 = reinterpret_cast<const v4f*>(y + rowbase);
    v4f*       ov4 = reinterpret_cast<v4f*>(out + rowbase);

    v4f yreg[PER_THREAD_VEC];

#if USE_WMMA
    v8f accx = {0.f, 0.f, 0.f, 0.f, 0.f, 0.f, 0.f, 0.f};
    v8f accy = {0.f, 0.f, 0.f, 0.f, 0.f, 0.f, 0.f, 0.f};
    const v2f ones = {1.0f, 1.0f};
#else
    float sx = 0.f, sy = 0.f;
#endif

#pragma unroll
    for (int j = 0; j < PER_THREAD_VEC; ++j) {
        const int vidx = j * BLOCK + tid;     // v4f index within the row (coalesced 4 KiB/iter)
        v4f xv = __builtin_nontemporal_load(&xv4[vidx]);
        v4f yv = __builtin_nontemporal_load(&yv4[vidx]);
        yreg[j] = yv;
#if USE_WMMA
        // Pairwise pre-add, then feed the matrix pipe: with B == ones(4x16),
        // D[m][n] += sum_k A[m][k]; every loaded value lands in A exactly once,
        // so sum(acc over all lanes/components) == 16 * (wave's element sum).
        v2f ax = { xv.x + xv.y, xv.z + xv.w };
        v2f ay = { yv.x + yv.y, yv.z + yv.w };
        accx = __builtin_amdgcn_wmma_f32_16x16x4_f32(
            false, ax, false, ones, (short)0, accx, false, false);
        accy = __builtin_amdgcn_wmma_f32_16x16x4_f32(
            false, ay, false, ones, (short)0, accy, false, false);
#else
        sx += xv.x + xv.y + xv.z + xv.w;
        sy += yv.x + yv.y + yv.z + yv.w;
#endif
    }

    // Per-lane partial sums.
#if USE_WMMA
    float px = ((accx[0] + accx[1]) + (accx[2] + accx[3])) +
               ((accx[4] + accx[5]) + (accx[6] + accx[7]));
    float py = ((accy[0] + accy[1]) + (accy[2] + accy[3])) +
               ((accy[4] + accy[5]) + (accy[6] + accy[7]));
    const float inv_div = 1.0f / (16.0f * (float)L_LEN);  // WMMA replicates across 16 cols
#else
    float px = sx, py = sy;
    const float inv_div = 1.0f / (float)L_LEN;
#endif

    // Block reduction (once per 8192-element row; cost negligible).
    __shared__ float redx[BLOCK];
    __shared__ float redy[BLOCK];
    redx[tid] = px;
    redy[tid] = py;
    __syncthreads();
#pragma unroll
    for (int s = BLOCK / 2; s > 0; s >>= 1) {
        if (tid < s) {
            redx[tid] += redx[tid + s];
            redy[tid] += redy[tid + s];
        }
        __syncthreads();
    }
    const float scale = (redx[0] + redy[0]) * inv_div;  // (sum_x + sum_y) / L

    // Scale the register-resident y and stream out non-temporally.
#pragma unroll
    for (int j = 0; j < PER_THREAD_VEC; ++j) {
        const int vidx = j * BLOCK + tid;
        v4f o;
        o.x = yreg[j].x * scale;
        o.y = yreg[j].y * scale;
        o.z = yreg[j].z * scale;
        o.w = yreg[j].w * scale;
        __builtin_nontemporal_store(o, &ov4[vidx]);
    }
}

extern "C" void kernel_launch(void* const* d_in, const int* in_sizes, int n_in,
                              void* d_out, int out_size, void* d_ws, size_t ws_size,
                              hipStream_t stream) {
    const float* x = (const float*)d_in[0];
    const float* y = (const float*)d_in[1];
    float* out = (float*)d_out;

    const int total = in_sizes[0];        // B*C*L = 64*64*8192
    const int rows  = total / L_LEN;      // 4096 rows of 8192 floats
    rowscale_kernel<<<dim3(rows), dim3(BLOCK), 0, stream>>>(x, y, out);
}